// MinBCWeightingModule_48198122996256
// MI455X (gfx1250) — compile-verified
//
#include <hip/hip_runtime.h>
#include <hip/hip_bf16.h>

#define NUM_TASKS   50
#define HSTRIDE     16                 // pad (task,source) bins to stride 16: idx = (t<<4)|s
#define NBINS       (NUM_TASKS * HSTRIDE)   // 800 words
#define NCOPY       4                  // privatized LDS histogram copies
#define HIST_BLOCKS 1024
#define GATH_BLOCKS 1024
#define THREADS     256                // 8 wave32 per block

__device__ __forceinline__ unsigned lds_off(const void* p) {
    // generic (flat) pointer to LDS: low 32 bits are the workgroup-relative LDS byte offset
    return (unsigned)(unsigned long long)p;
}

__global__ void zero_bins_kernel(unsigned* __restrict__ bins) {
    for (int i = threadIdx.x; i < NBINS; i += blockDim.x) bins[i] = 0u;
}

// ---------------------------------------------------------------------------
// Pass 1: (task,source) histogram.
// Streams task_ids / source_ids tiles into LDS with global_load_async_to_lds_b128
// (ASYNCcnt double-buffering), bumps privatized LDS histograms, merges to global.
// Tile = 256 threads * 4 ints = 1024 elements per array per block-step.
// ---------------------------------------------------------------------------
__global__ void hist_kernel(const int* __restrict__ t, const int* __restrict__ s,
                            unsigned* __restrict__ gbins, long long n) {
    __shared__ unsigned hc[NCOPY][NBINS];
    __shared__ int ldsT[2][THREADS * 4];
    __shared__ int ldsS[2][THREADS * 4];

    const int tid = threadIdx.x;
    for (int i = tid; i < NCOPY * NBINS; i += blockDim.x) (&hc[0][0])[i] = 0u;
    __syncthreads();

    unsigned* myh = hc[(tid >> 5) & (NCOPY - 1)];   // 2 waves share a copy

    const long long nv4   = n >> 2;          // int4 count
    const long long tiles = nv4 >> 8;        // full 1024-elem tiles
    const long long gstep = gridDim.x;

    const unsigned laT0 = lds_off(&ldsT[0][tid << 2]);
    const unsigned laT1 = lds_off(&ldsT[1][tid << 2]);
    const unsigned laS0 = lds_off(&ldsS[0][tid << 2]);
    const unsigned laS1 = lds_off(&ldsS[1][tid << 2]);

#define ISSUE_TILE(TI, B)                                                         \
    do {                                                                          \
        long long _e0 = ((long long)(TI) << 10) + ((long long)tid << 2);          \
        unsigned long long _gaT = (unsigned long long)(t + _e0);                  \
        unsigned long long _gaS = (unsigned long long)(s + _e0);                  \
        unsigned _laT = (B) ? laT1 : laT0;                                        \
        unsigned _laS = (B) ? laS1 : laS0;                                        \
        asm volatile("global_load_async_to_lds_b128 %0, %1, off"                  \
                     :: "v"(_laT), "v"(_gaT) : "memory");                         \
        asm volatile("global_load_async_to_lds_b128 %0, %1, off"                  \
                     :: "v"(_laS), "v"(_gaS) : "memory");                         \
    } while (0)

    long long tile0 = blockIdx.x;
    if (tile0 < tiles)          ISSUE_TILE(tile0, 0);
    if (tile0 + gstep < tiles)  ISSUE_TILE(tile0 + gstep, 1);

    int buf = 0;
    for (long long k = tile0; k < tiles; k += gstep) {
        if (k + gstep < tiles)
            asm volatile("s_wait_asynccnt 0x2" ::: "memory");  // tile k done, k+1 in flight
        else
            asm volatile("s_wait_asynccnt 0x0" ::: "memory");  // drain

        int4 tv = *(const int4*)&ldsT[buf][tid << 2];
        int4 sv = *(const int4*)&ldsS[buf][tid << 2];
        atomicAdd(&myh[(tv.x << 4) + sv.x], 1u);
        atomicAdd(&myh[(tv.y << 4) + sv.y], 1u);
        atomicAdd(&myh[(tv.z << 4) + sv.z], 1u);
        atomicAdd(&myh[(tv.w << 4) + sv.w], 1u);

        long long nk = k + 2 * gstep;
        if (nk < tiles) {
            asm volatile("s_wait_dscnt 0x0" ::: "memory");  // buffer reads done before reuse
            ISSUE_TILE(nk, buf);
        }
        buf ^= 1;
    }
#undef ISSUE_TILE

    // scalar tail (n not covered by full tiles; empty for 2^23 elements)
    {
        long long done    = tiles << 10;
        long long gid     = (long long)blockIdx.x * blockDim.x + tid;
        long long gstride = (long long)gridDim.x * blockDim.x;
        for (long long j = done + gid; j < n; j += gstride)
            atomicAdd(&myh[(t[j] << 4) + s[j]], 1u);
    }

    __syncthreads();
    for (int b = tid; b < NBINS; b += blockDim.x) {
        unsigned sum = hc[0][b] + hc[1][b] + hc[2][b] + hc[3][b];
        if (sum) atomicAdd(&gbins[b], sum);
    }
}

// ---------------------------------------------------------------------------
// Pass 2: 550-entry weight table. One block, one thread per task.
// w[t][s] = (cnt_t / N) * exp(l_ts - max_t) / denom_t   (mirrors reference f32 ops)
// ---------------------------------------------------------------------------
__global__ void weights_kernel(const unsigned* __restrict__ bins,
                               const float* __restrict__ wl,
                               float* __restrict__ wtab, long long n) {
    int task = threadIdx.x;
    if (task >= NUM_TASKS) return;

    float    l[11];
    unsigned c[11];
    for (int sc = 0; sc < 11; ++sc) {
        l[sc] = wl[task * 11 + sc];
        c[sc] = bins[(task << 4) + sc];
    }
    float maxv = -3.402823466e38f;
    unsigned long long cnt = 0;
    for (int sc = 1; sc <= 10; ++sc) {
        cnt += c[sc];
        if (c[sc] && l[sc] > maxv) maxv = l[sc];
    }
    float denom = 0.0f;
    for (int sc = 1; sc <= 10; ++sc)
        if (c[sc]) denom += (float)c[sc] * expf(l[sc] - maxv);

    float cfrac = (float)cnt / (float)n;
    for (int sc = 0; sc < 11; ++sc) {
        float w = 0.0f;
        if (cnt > 0) {
            float e = expf(l[sc] - maxv);
            w = cfrac * (e / denom);
        }
        wtab[(task << 4) + sc] = w;
    }
}

// ---------------------------------------------------------------------------
// Pass 3: gather out[i] = w[t_i*16 + s_i].  int4 loads / float4 stores.
// Pass-1 already pulled t/s into the 192MB L2, so reads come from L2.
// ---------------------------------------------------------------------------
__global__ void gather_kernel(const int* __restrict__ t, const int* __restrict__ s,
                              const float* __restrict__ wtab, float* __restrict__ out,
                              long long n) {
    __shared__ float w[NBINS];
    for (int i = threadIdx.x; i < NBINS; i += blockDim.x) w[i] = wtab[i];
    __syncthreads();

    const long long nv4     = n >> 2;
    long long       i       = (long long)blockIdx.x * blockDim.x + threadIdx.x;
    const long long stride  = (long long)gridDim.x * blockDim.x;
    const int4*  t4 = (const int4*)t;
    const int4*  s4 = (const int4*)s;
    float4*      o4 = (float4*)out;

    for (; i < nv4; i += stride) {
        int4 tv = t4[i], sv = s4[i];
        float4 r;
        r.x = w[(tv.x << 4) + sv.x];
        r.y = w[(tv.y << 4) + sv.y];
        r.z = w[(tv.z << 4) + sv.z];
        r.w = w[(tv.w << 4) + sv.w];
        o4[i] = r;
    }
    // scalar tail
    long long gid = (long long)blockIdx.x * blockDim.x + threadIdx.x;
    for (long long j = (nv4 << 2) + gid; j < n; j += stride)
        out[j] = w[(t[j] << 4) + s[j]];
}

extern "C" void kernel_launch(void* const* d_in, const int* in_sizes, int n_in,
                              void* d_out, int out_size, void* d_ws, size_t ws_size,
                              hipStream_t stream) {
    const int*   task = (const int*)d_in[0];
    // d_in[1] (variant_ids) is unused by the reference.
    const int*   src  = (const int*)d_in[2];
    const float* wl   = (const float*)d_in[3];
    const long long n = (long long)in_sizes[0];

    unsigned* bins = (unsigned*)d_ws;            // NBINS u32
    float*    wtab = (float*)d_ws + NBINS;       // NBINS f32

    zero_bins_kernel<<<1, THREADS, 0, stream>>>(bins);
    hist_kernel<<<HIST_BLOCKS, THREADS, 0, stream>>>(task, src, bins, n);
    weights_kernel<<<1, 64, 0, stream>>>(bins, wl, wtab, n);
    gather_kernel<<<GATH_BLOCKS, THREADS, 0, stream>>>(task, src, wtab, (float*)d_out, n);
}